// ProgressiveModel_45200235823175
// MI455X (gfx1250) — compile-verified
//
#include <hip/hip_runtime.h>
#include <math.h>

#define D_MODEL   512
#define D_STATE   32
#define HEADDIM   64
#define D_INNER   1024
#define NHEADS    16
#define D_CONV    4
#define CONV_DIM  1088      // D_INNER + 2*D_STATE
#define D_IN_PROJ 2128      // 2*D_INNER + 2*D_STATE + NHEADS
#define NB        2
#define NL        1024
#define NTOK      2048      // NB*NL
#define VOCABSZ   260
#define D_ATTN    64

typedef float v2f __attribute__((ext_vector_type(2)));
typedef float v8f __attribute__((ext_vector_type(8)));

// ---------------- block reduction helpers (blockDim.x == 256, 8 wave32s) ----
__device__ __forceinline__ float blk_sum(float v, volatile float* sh) {
  int t = threadIdx.x;
#pragma unroll
  for (int o = 16; o; o >>= 1) v += __shfl_xor(v, o, 32);
  __syncthreads();
  if ((t & 31) == 0) sh[t >> 5] = v;
  __syncthreads();
  float r = sh[0];
#pragma unroll
  for (int i = 1; i < 8; ++i) r += sh[i];
  return r;
}

__device__ __forceinline__ float blk_max(float v, volatile float* sh) {
  int t = threadIdx.x;
#pragma unroll
  for (int o = 16; o; o >>= 1) v = fmaxf(v, __shfl_xor(v, o, 32));
  __syncthreads();
  if ((t & 31) == 0) sh[t >> 5] = v;
  __syncthreads();
  float r = sh[0];
#pragma unroll
  for (int i = 1; i < 8; ++i) r = fmaxf(r, sh[i]);
  return r;
}

__device__ __forceinline__ float silu(float x) { return x / (1.f + expf(-x)); }

// ---------------- LayerNorm kernels ----------------------------------------
__global__ __launch_bounds__(256)
void k_ln(const float* __restrict__ in, const float* __restrict__ g,
          const float* __restrict__ b, float* __restrict__ out) {
  __shared__ float sh[8];
  int row = blockIdx.x;
  const float* xr = in + (size_t)row * D_MODEL;
  float s = 0.f, s2 = 0.f;
  for (int j = threadIdx.x; j < D_MODEL; j += 256) {
    float v = xr[j]; s += v; s2 += v * v;
  }
  s  = blk_sum(s, sh);
  s2 = blk_sum(s2, sh);
  float mu = s * (1.f / D_MODEL);
  float var = s2 * (1.f / D_MODEL) - mu * mu;
  float rs = rsqrtf(var + 1e-5f);
  for (int j = threadIdx.x; j < D_MODEL; j += 256)
    out[(size_t)row * D_MODEL + j] = (xr[j] - mu) * rs * g[j] + b[j];
}

__global__ __launch_bounds__(256)
void k_embed_ln(const int* __restrict__ tok, const float* __restrict__ emb,
                const float* __restrict__ g, const float* __restrict__ b,
                float* __restrict__ out) {
  __shared__ float sh[8];
  int row = blockIdx.x;
  const float* xr = emb + (size_t)tok[row] * D_MODEL;
  float s = 0.f, s2 = 0.f;
  for (int j = threadIdx.x; j < D_MODEL; j += 256) {
    float v = xr[j]; s += v; s2 += v * v;
  }
  s  = blk_sum(s, sh);
  s2 = blk_sum(s2, sh);
  float mu = s * (1.f / D_MODEL);
  float var = s2 * (1.f / D_MODEL) - mu * mu;
  float rs = rsqrtf(var + 1e-5f);
  for (int j = threadIdx.x; j < D_MODEL; j += 256)
    out[(size_t)row * D_MODEL + j] = (xr[j] - mu) * rs * g[j] + b[j];
}

// ---------------- generic WMMA f32 GEMM ------------------------------------
// C[M,N] = alpha * A[M,K] @ W (+ bias[n]); W element (k,n) at W[k*WSK + n*WSN]
// WSK/WSN are compile-time so B addresses fold into immediate offsets.
// block = 128 threads (4 wave32s in 2x2); wave tile 32x32 (4 WMMA accs);
// block tile 64x64. grid: ((N+63)/64, M/64). M must be a multiple of 64.
// Branch-free N edge: clamped-address loads scaled by a 0/1 mask.
template <int WSK, int WSN>
__global__ __launch_bounds__(128)
void k_gemm(const float* __restrict__ A, const float* __restrict__ W,
            const float* __restrict__ bias, float* __restrict__ C,
            int M, int N, int K, int lda, int ldc, float alpha) {
  int lane = threadIdx.x & 31;
  int wv   = threadIdx.x >> 5;            // 0..3
  int wr   = wv >> 1;                     // 2 wave-rows
  int wc   = wv & 1;                      // 2 wave-cols
  int row0 = blockIdx.y * 64 + wr * 32;
  int col0 = blockIdx.x * 64 + wc * 32;
  if (col0 >= N) return;                  // wave-uniform dead-tile exit

  int mn    = lane & 15;                  // M index (A) / N index (B,C)
  int khalf = (lane >> 4) * 2;            // K sub-offset: lanes 16-31 -> +2

  int c0 = col0 + mn;
  int c1 = c0 + 16;
  float msk0 = (c0 < N) ? 1.f : 0.f;
  float msk1 = (c1 < N) ? 1.f : 0.f;
  int cc0 = (c0 < N) ? c0 : (N - 1);      // clamped (always valid) columns
  int cc1 = (c1 < N) ? c1 : (N - 1);

  const float* pa0 = A + (size_t)(row0 + mn) * lda + khalf;
  const float* pa1 = pa0 + (size_t)16 * lda;
  const float* pb0 = W + (size_t)khalf * WSK + (size_t)cc0 * WSN;
  const float* pb1 = W + (size_t)khalf * WSK + (size_t)cc1 * WSN;

  v8f acc00 = {0.f,0.f,0.f,0.f,0.f,0.f,0.f,0.f};
  v8f acc01 = acc00, acc10 = acc00, acc11 = acc00;

#pragma unroll 4
  for (int k0 = 0; k0 < K; k0 += 4) {
    v2f a0, a1, b0, b1;
    a0.x = pa0[0]; a0.y = pa0[1];
    a1.x = pa1[0]; a1.y = pa1[1];
    b0.x = pb0[0] * msk0; b0.y = pb0[WSK] * msk0;
    b1.x = pb1[0] * msk1; b1.y = pb1[WSK] * msk1;
    acc00 = __builtin_amdgcn_wmma_f32_16x16x4_f32(false, a0, false, b0,
                                                  (short)0, acc00, false, false);
    acc01 = __builtin_amdgcn_wmma_f32_16x16x4_f32(false, a0, false, b1,
                                                  (short)0, acc01, false, false);
    acc10 = __builtin_amdgcn_wmma_f32_16x16x4_f32(false, a1, false, b0,
                                                  (short)0, acc10, false, false);
    acc11 = __builtin_amdgcn_wmma_f32_16x16x4_f32(false, a1, false, b1,
                                                  (short)0, acc11, false, false);
    pa0 += 4; pa1 += 4;
    pb0 += 4 * WSK; pb1 += 4 * WSK;
  }

  bool ok0 = c0 < N, ok1 = c1 < N;
  float bv0 = (bias != nullptr && ok0) ? bias[c0] : 0.f;
  float bv1 = (bias != nullptr && ok1) ? bias[c1] : 0.f;
  int m0 = row0 + ((lane >> 4) << 3);     // C rows: r + (lanehi ? 8 : 0)
#pragma unroll
  for (int r = 0; r < 8; ++r) {
    if (ok0) C[(size_t)(m0 + r) * ldc + c0]      = alpha * acc00[r] + bv0;
    if (ok1) C[(size_t)(m0 + r) * ldc + c1]      = alpha * acc01[r] + bv1;
    if (ok0) C[(size_t)(m0 + 16 + r) * ldc + c0] = alpha * acc10[r] + bv0;
    if (ok1) C[(size_t)(m0 + 16 + r) * ldc + c1] = alpha * acc11[r] + bv1;
  }
}

// ---------------- causal conv + SiLU over xBC ------------------------------
__global__ __launch_bounds__(256)
void k_conv(const float* __restrict__ zx, const float* __restrict__ cw,
            const float* __restrict__ cb, float* __restrict__ out) {
  int idx = blockIdx.x * 256 + threadIdx.x;
  if (idx >= NTOK * CONV_DIM) return;
  int c   = idx % CONV_DIM;
  int row = idx / CONV_DIM;               // b*NL + l
  int l   = row & (NL - 1);
  float acc = cb[c];
#pragma unroll
  for (int k = 0; k < D_CONV; ++k) {
    int ls = l - (D_CONV - 1) + k;
    if (ls >= 0)
      acc += cw[k * CONV_DIM + c] *
             zx[(size_t)(row - (D_CONV - 1) + k) * D_IN_PROJ + D_INNER + c];
  }
  out[(size_t)row * CONV_DIM + c] = silu(acc);
}

// ---------------- dt = softplus(dt_raw + bias), decay = exp(-exp(A_log)*dt) -
__global__ __launch_bounds__(256)
void k_dt(const float* __restrict__ zx, const float* __restrict__ dtb,
          const float* __restrict__ Alog, float* __restrict__ dtv,
          float* __restrict__ dec) {
  int idx = blockIdx.x * 256 + threadIdx.x;
  if (idx >= NTOK * NHEADS) return;
  int h   = idx % NHEADS;
  int row = idx / NHEADS;
  float d = zx[(size_t)row * D_IN_PROJ + (D_INNER + CONV_DIM) + h] + dtb[h];
  float sp = (d > 20.f) ? d : log1pf(expf(d));
  dtv[idx] = sp;
  dec[idx] = expf(-expf(Alog[h]) * sp);
}

// ---------------- selective-state-space scan -------------------------------
// one block per (b,h); 256 threads; thread t owns p = t>>2, n = (t&3)*8 .. +8
__global__ __launch_bounds__(256)
void k_scan(const float* __restrict__ xconv, const float* __restrict__ dtv,
            const float* __restrict__ dec, const float* __restrict__ zx,
            const float* __restrict__ Dssm, float* __restrict__ y) {
  int b = blockIdx.x >> 4;
  int h = blockIdx.x & 15;
  int t = threadIdx.x;
  int p  = t >> 2;
  int nb = (t & 3) << 3;
  float s[8] = {0.f, 0.f, 0.f, 0.f, 0.f, 0.f, 0.f, 0.f};
  __shared__ float shx[HEADDIM];
  __shared__ float shB[D_STATE];
  __shared__ float shC[D_STATE];
  float Dh = Dssm[h];

  for (int tt = 0; tt < NL; ++tt) {
    int row = b * NL + tt;
    const float* xr = xconv + (size_t)row * CONV_DIM;
    if (t < 64)       shx[t]      = xr[h * HEADDIM + t];
    else if (t < 96)  shB[t - 64] = xr[D_INNER + (t - 64)];
    else if (t < 128) shC[t - 96] = xr[D_INNER + D_STATE + (t - 96)];
    __syncthreads();

    float a = dec[(size_t)row * NHEADS + h];
    float d = dtv[(size_t)row * NHEADS + h];
    float du = d * shx[p];
    float part = 0.f;
#pragma unroll
    for (int j = 0; j < 8; ++j) {
      s[j] = a * s[j] + du * shB[nb + j];
      part += s[j] * shC[nb + j];
    }
    part += __shfl_xor(part, 1, 32);
    part += __shfl_xor(part, 2, 32);
    if ((t & 3) == 0) {
      float val = part + Dh * shx[p];
      float z = zx[(size_t)row * D_IN_PROJ + h * HEADDIM + p];
      y[(size_t)row * D_INNER + h * HEADDIM + p] = val * silu(z);
    }
    __syncthreads();
  }
}

// ---------------- residual / mix updates -----------------------------------
__global__ __launch_bounds__(256)
void k_axpy(float* __restrict__ x, const float* __restrict__ hadd,
            const float* __restrict__ scale, int layer, int n) {
  int i = blockIdx.x * 256 + threadIdx.x;
  if (i < n) x[i] += scale[layer] * hadd[i];
}

// ---------------- causal softmax over score rows ---------------------------
__global__ __launch_bounds__(256)
void k_softmax(float* __restrict__ sc) {
  __shared__ float sh[8];
  int row = blockIdx.x;                   // 0..NTOK-1
  int l   = row & (NL - 1);               // causal limit (inclusive)
  float* sr = sc + (size_t)row * NL;
  int t = threadIdx.x;
  float m = -1e30f;
  for (int j = t; j <= l; j += 256) m = fmaxf(m, sr[j]);
  m = blk_max(m, sh);
  float sum = 0.f;
  for (int j = t; j < NL; j += 256) {
    float v = (j <= l) ? expf(sr[j] - m) : 0.f;
    sr[j] = v;
    sum += v;
  }
  sum = blk_sum(sum, sh);
  float inv = 1.f / sum;
  for (int j = t; j < NL; j += 256) sr[j] *= inv;
}

// ---------------------------------------------------------------------------
extern "C" void kernel_launch(void* const* d_in, const int* in_sizes, int n_in,
                              void* d_out, int out_size, void* d_ws,
                              size_t ws_size, hipStream_t stream) {
  const int*   tokens   = (const int*)  d_in[0];
  const float* embed_W  = (const float*)d_in[1];
  const float* eng      = (const float*)d_in[2];
  const float* enb      = (const float*)d_in[3];
  const float* norm_g   = (const float*)d_in[4];
  const float* norm_b   = (const float*)d_in[5];
  const float* lscale   = (const float*)d_in[6];
  const float* in_W     = (const float*)d_in[7];
  const float* conv_W   = (const float*)d_in[8];
  const float* conv_b   = (const float*)d_in[9];
  const float* dt_bias  = (const float*)d_in[10];
  const float* A_log    = (const float*)d_in[11];
  const float* D_ssm    = (const float*)d_in[12];
  const float* out_W    = (const float*)d_in[13];
  const float* q_W      = (const float*)d_in[14];
  const float* q_b      = (const float*)d_in[15];
  const float* k_W      = (const float*)d_in[16];
  const float* k_b      = (const float*)d_in[17];
  const float* v_W      = (const float*)d_in[18];
  const float* v_b      = (const float*)d_in[19];
  const float* o_W      = (const float*)d_in[20];
  const float* o_b      = (const float*)d_in[21];
  const float* mix      = (const float*)d_in[22];
  const float* fng      = (const float*)d_in[23];
  const float* fnb      = (const float*)d_in[24];
  float* logits = (float*)d_out;

  float* ws = (float*)d_ws;
  size_t o = 0;
  float* x     = ws + o; o += (size_t)NTOK * D_MODEL;
  float* xn    = ws + o; o += (size_t)NTOK * D_MODEL;
  float* zx    = ws + o; o += (size_t)NTOK * D_IN_PROJ;
  float* xconv = ws + o; o += (size_t)NTOK * CONV_DIM;
  float* dtv   = ws + o; o += (size_t)NTOK * NHEADS;
  float* dec   = ws + o; o += (size_t)NTOK * NHEADS;
  float* yssm  = ws + o; o += (size_t)NTOK * D_INNER;
  float* hout  = ws + o; o += (size_t)NTOK * D_MODEL;
  float* Q     = ws + o; o += (size_t)NTOK * D_ATTN;
  float* Km    = ws + o; o += (size_t)NTOK * D_ATTN;
  float* V     = ws + o; o += (size_t)NTOK * D_MODEL;
  float* sc    = ws + o; o += (size_t)NB * NL * NL;
  float* att   = ws + o; o += (size_t)NTOK * D_MODEL;
  float* Ob    = ws + o; o += (size_t)NTOK * D_MODEL;

  dim3 blk(256);
  dim3 gblk(128);

  // embed + LN
  k_embed_ln<<<NTOK, blk, 0, stream>>>(tokens, embed_W, eng, enb, x);

  for (int i = 0; i < 4; ++i) {
    k_ln<<<NTOK, blk, 0, stream>>>(x, norm_g + i * D_MODEL,
                                   norm_b + i * D_MODEL, xn);
    // zxbcdt = xn @ in_proj_W[i]   (2048 x 512 x 2128)
    k_gemm<D_IN_PROJ, 1><<<dim3((D_IN_PROJ + 63) / 64, NTOK / 64), gblk, 0,
                           stream>>>(
        xn, in_W + (size_t)i * D_MODEL * D_IN_PROJ, nullptr, zx,
        NTOK, D_IN_PROJ, D_MODEL, D_MODEL, D_IN_PROJ, 1.f);
    // causal conv + silu
    k_conv<<<(NTOK * CONV_DIM + 255) / 256, blk, 0, stream>>>(
        zx, conv_W + (size_t)i * D_CONV * CONV_DIM,
        conv_b + (size_t)i * CONV_DIM, xconv);
    // dt / decay
    k_dt<<<(NTOK * NHEADS + 255) / 256, blk, 0, stream>>>(
        zx, dt_bias + i * NHEADS, A_log + i * NHEADS, dtv, dec);
    // selective scan
    k_scan<<<NB * NHEADS, blk, 0, stream>>>(xconv, dtv, dec, zx,
                                            D_ssm + i * NHEADS, yssm);
    // hout = yssm @ out_proj_W[i]   (2048 x 1024 x 512)
    k_gemm<D_MODEL, 1><<<dim3(D_MODEL / 64, NTOK / 64), gblk, 0, stream>>>(
        yssm, out_W + (size_t)i * D_INNER * D_MODEL, nullptr, hout,
        NTOK, D_MODEL, D_INNER, D_INNER, D_MODEL, 1.f);
    // x += layer_scale[i] * hout
    k_axpy<<<(NTOK * D_MODEL + 255) / 256, blk, 0, stream>>>(
        x, hout, lscale, i, NTOK * D_MODEL);
  }

  // ---- history attention ----
  k_gemm<D_ATTN, 1><<<dim3(1, NTOK / 64), gblk, 0, stream>>>(
      x, q_W, q_b, Q, NTOK, D_ATTN, D_MODEL, D_MODEL, D_ATTN, 1.f);
  k_gemm<D_ATTN, 1><<<dim3(1, NTOK / 64), gblk, 0, stream>>>(
      x, k_W, k_b, Km, NTOK, D_ATTN, D_MODEL, D_MODEL, D_ATTN, 1.f);
  k_gemm<D_MODEL, 1><<<dim3(D_MODEL / 64, NTOK / 64), gblk, 0, stream>>>(
      x, v_W, v_b, V, NTOK, D_MODEL, D_MODEL, D_MODEL, D_MODEL, 1.f);

  for (int b = 0; b < NB; ++b) {
    // scores_b = (Q_b @ K_b^T) / 8   (1024 x 64 x 1024); W strided as K^T
    k_gemm<1, D_ATTN><<<dim3(NL / 64, NL / 64), gblk, 0, stream>>>(
        Q + (size_t)b * NL * D_ATTN, Km + (size_t)b * NL * D_ATTN, nullptr,
        sc + (size_t)b * NL * NL, NL, NL, D_ATTN, D_ATTN, NL, 0.125f);
  }
  k_softmax<<<NTOK, blk, 0, stream>>>(sc);
  for (int b = 0; b < NB; ++b) {
    // att_b = w_b @ V_b   (1024 x 1024 x 512)
    k_gemm<D_MODEL, 1><<<dim3(D_MODEL / 64, NL / 64), gblk, 0, stream>>>(
        sc + (size_t)b * NL * NL, V + (size_t)b * NL * D_MODEL, nullptr,
        att + (size_t)b * NL * D_MODEL, NL, D_MODEL, NL, NL, D_MODEL, 1.f);
  }
  // O = att @ o_W + o_b
  k_gemm<D_MODEL, 1><<<dim3(D_MODEL / 64, NTOK / 64), gblk, 0, stream>>>(
      att, o_W, o_b, Ob, NTOK, D_MODEL, D_MODEL, D_MODEL, D_MODEL, 1.f);
  // x += mix * O
  k_axpy<<<(NTOK * D_MODEL + 255) / 256, blk, 0, stream>>>(
      x, Ob, mix, 0, NTOK * D_MODEL);

  // final LN + logits = xn @ embed_W^T  (W strided: wsk=1, wsn=512, N=260)
  k_ln<<<NTOK, blk, 0, stream>>>(x, fng, fnb, xn);
  k_gemm<1, D_MODEL><<<dim3((VOCABSZ + 63) / 64, NTOK / 64), gblk, 0,
                       stream>>>(
      xn, embed_W, nullptr, logits, NTOK, VOCABSZ, D_MODEL, D_MODEL,
      VOCABSZ, 1.f);
}